// D3PM_81466939671320
// MI455X (gfx1250) — compile-verified
//
#include <hip/hip_runtime.h>
#include <hip/hip_bf16.h>

#define KDIM   8193
#define VOCABN 8192
#define TSTEPS 1000
#define NROWS  8192      // B*L = 8*1024
#define EPSF   1e-20f
#define LAMCE  0.01f

typedef float v4f __attribute__((ext_vector_type(4)));
typedef float v2f __attribute__((ext_vector_type(2)));
typedef float v8f __attribute__((ext_vector_type(8)));

// -------- wave32 sum reduction via V_WMMA_F32_16X16X4_F32 (ones trick) --------
// A[m,0]=v[m], A[m,2]=v[m+16] (A vgpr0=v, vgpr1=0 per 16x4 f32 layout), B=ones.
// D[m,n] = v[m] + v[m+16]. Lane<16 holds rows 0..7, lane>=16 rows 8..15 in d[0..7].
// Sum d[0..7] locally, then one xor-16 shuffle gives the full 32-lane sum in all lanes.
__device__ __forceinline__ float wave_sum32(float v) {
#if __has_builtin(__builtin_amdgcn_wmma_f32_16x16x4_f32)
    v2f a; a.x = v;    a.y = 0.0f;
    v2f b; b.x = 1.0f; b.y = 1.0f;
    v8f c = {};
    v8f d = __builtin_amdgcn_wmma_f32_16x16x4_f32(false, a, false, b, (short)0, c, false, false);
    float h = ((d[0] + d[1]) + (d[2] + d[3])) + ((d[4] + d[5]) + (d[6] + d[7]));
    h += __shfl_xor(h, 16, 32);
    return h;
#else
    #pragma unroll
    for (int off = 16; off; off >>= 1) v += __shfl_xor(v, off, 32);
    return v;
#endif
}

// unnormalized posterior factors at a special index k (t > 0 path)
__device__ __forceinline__ void uq_up_at(int k, int i, int j, int Mi, bool jM,
                                         float abar, float beta1, float pk, float sp,
                                         float& uq, float& up) {
    float A  = jM ? (k == Mi ? 1.0f : beta1) : (k == j ? (1.0f - beta1) : 0.0f);
    float Bq = (k == i ? abar : 0.0f) + (k == Mi ? (1.0f - abar) : 0.0f);
    float Bp = abar * pk + (k == Mi ? (1.0f - abar) * sp : 0.0f);
    uq = (A + EPSF) * (Bq + EPSF);
    up = (A + EPSF) * (Bp + EPSF);
}

__global__ __launch_bounds__(256) void d3pm_row_kernel(
    const float* __restrict__ logits, const int* __restrict__ data,
    const int* __restrict__ tarr, const int* __restrict__ xtp1,
    float* __restrict__ out_logits, float* __restrict__ lvb_ws, float* __restrict__ lce_ws)
{
    const int r   = blockIdx.x;           // row id in [0, 8192)
    const int tid = threadIdx.x;
    const size_t base = (size_t)r * KDIM;
    const float* __restrict__ row  = logits + base;
    float* __restrict__       orow = out_logits + base;

    // ---- alignment peel (row stride 8193 floats => base misaligned by r%4 floats)
    const int mis  = (int)(base & 3);
    const int head = (4 - mis) & 3;
    const int n4   = (KDIM - head) >> 2;          // 2047 or 2048 float4 chunks
    const int tb   = head + (n4 << 2);
    const int tail = KDIM - tb;                   // 0..3

    // thread 0: fetch indices early + prefetch the 3 gather targets
    int ii = 0, jj = 0, tt = 0;
    if (tid == 0) {
        ii = data[r]; jj = xtp1[r]; tt = tarr[r >> 10];
        __builtin_prefetch(row + ii, 0, 3);
        __builtin_prefetch(row + jj, 0, 3);
        __builtin_prefetch(row + (KDIM - 1), 0, 3);
    }

    // ---- pass 1: streaming load (NT), fused echo copy (NT), thread-local max
    float lmax = -INFINITY;
    v4f vals[8];
    const v4f* __restrict__ row4  = (const v4f*)(row + head);
    v4f* __restrict__       orow4 = (v4f*)(orow + head);
    #pragma unroll
    for (int u = 0; u < 8; ++u) {
        int c = tid + (u << 8);
        if (c < n4) {
            v4f v = __builtin_nontemporal_load(row4 + c);
            vals[u] = v;
            __builtin_nontemporal_store(v, orow4 + c);
            lmax = fmaxf(lmax, fmaxf(fmaxf(v.x, v.y), fmaxf(v.z, v.w)));
        }
    }
    float extra[6]; int ne = 0;
    if (tid == 255) {                              // head + tail scalars (<=5 total)
        for (int e = 0; e < head; ++e) { float x = row[e];    extra[ne++] = x; orow[e]    = x; lmax = fmaxf(lmax, x); }
        for (int e = 0; e < tail; ++e) { float x = row[tb+e]; extra[ne++] = x; orow[tb+e] = x; lmax = fmaxf(lmax, x); }
    }

    // ---- block max
    __shared__ float red[256];
    __shared__ float sbc;
    red[tid] = lmax;
    __syncthreads();
    if (tid < 32) {
        float m8 = red[tid];
        #pragma unroll
        for (int w = 1; w < 8; ++w) m8 = fmaxf(m8, red[tid + (w << 5)]);
        #pragma unroll
        for (int off = 16; off; off >>= 1) m8 = fmaxf(m8, __shfl_xor(m8, off, 32));
        if (tid == 0) sbc = m8;
    }
    __syncthreads();
    const float m = sbc;

    // ---- pass 2 (registers): sum exp(x - m)
    float lz = 0.0f;
    #pragma unroll
    for (int u = 0; u < 8; ++u) {
        int c = tid + (u << 8);
        if (c < n4) {
            v4f v = vals[u];
            lz += (__expf(v.x - m) + __expf(v.y - m)) + (__expf(v.z - m) + __expf(v.w - m));
        }
    }
    if (tid == 255) for (int e = 0; e < ne; ++e) lz += __expf(extra[e] - m);
    __syncthreads();
    red[tid] = lz;
    __syncthreads();
    if (tid < 32) {
        float s8 = red[tid];
        #pragma unroll
        for (int w = 1; w < 8; ++w) s8 += red[tid + (w << 5)];
        float Zs = wave_sum32(s8);                 // v_wmma_f32_16x16x4_f32
        if (tid == 0) sbc = Zs;
    }
    __syncthreads();
    const float Z = sbc;

    // ---- O(1) epilogue: the posterior has <=3 non-trivial indices (i, j, mask)
    if (tid == 0) {
        const int Mi = KDIM - 1;
        float logZ = __logf(Z);
        float invZ = 1.0f / Z;
        float xi   = row[ii];
        float l_ce = (m + logZ) - xi;              // -log_softmax(logits)[data]
        float p_i  = __expf(xi - m) * invZ;
        float lvb;
        if (tt == 0) {
            // lq ~ delta_i; l_vb = (log(1+eps)-logZq0) - (log(p_i+eps)-logZp0)
            float Zq0 = (1.0f + EPSF) + (float)(KDIM - 1) * EPSF;
            float Zp0 = 1.0f + (float)KDIM * EPSF;
            float a = (__logf(1.0f + EPSF) - __logf(Zq0))
                    - (__logf(p_i + EPSF) - __logf(Zp0));
            lvb = fmaxf(a, 0.0f);
        } else {
            float abar  = (float)(TSTEPS - tt) / (float)TSTEPS;   // cumprod telescopes
            float beta1 = 1.0f / (float)(TSTEPS - tt);
            float xj  = row[jj], xM = row[Mi];
            float p_j = __expf(xj - m) * invZ;
            float p_M = __expf(xM - m) * invZ;
            const float sp = 1.0f;                                 // sum(softmax)
            bool jM = (jj == Mi), ji = (jj == ii);

            float uq0, up0, uq1, up1, uq2 = 0.0f, up2 = 0.0f;
            uq_up_at(ii, ii, jj, Mi, jM, abar, beta1, p_i, sp, uq0, up0);
            uq_up_at(Mi, ii, jj, Mi, jM, abar, beta1, p_M, sp, uq1, up1);
            float psum = p_i + p_M;
            int nspec = 2;
            if (!ji && !jM) {
                uq_up_at(jj, ii, jj, Mi, jM, abar, beta1, p_j, sp, uq2, up2);
                nspec = 3; psum += p_j;
            }
            float n_other = (float)(KDIM - nspec);
            float Aother  = jM ? beta1 : 0.0f;
            float Zq = (uq0 + uq1 + uq2) + n_other * ((Aother + EPSF) * EPSF);
            float p_rest = fmaxf(sp - psum, 0.0f);
            float Zp = (up0 + up1 + up2) + (Aother + EPSF) * (abar * p_rest + n_other * EPSF);
            float lZq = __logf(Zq), lZp = __logf(Zp);
            float acc = (uq0 / Zq) * ((__logf(uq0) - lZq) - (__logf(up0) - lZp))
                      + (uq1 / Zq) * ((__logf(uq1) - lZq) - (__logf(up1) - lZp));
            if (nspec == 3)
                acc += (uq2 / Zq) * ((__logf(uq2) - lZq) - (__logf(up2) - lZp));
            lvb = fmaxf(acc, 0.0f);
        }
        lvb_ws[r] = lvb;
        lce_ws[r] = l_ce;
    }
}

__global__ __launch_bounds__(256) void d3pm_final_kernel(
    const float* __restrict__ lvb_ws, const float* __restrict__ lce_ws,
    const int* __restrict__ tarr, float* __restrict__ out5)
{
    const int tid = threadIdx.x;
    float svb = 0.0f, sce = 0.0f, sloss = 0.0f, sce0 = 0.0f;
    for (int r = tid; r < NROWS; r += 256) {
        int tt = tarr[r >> 10];
        float lv = lvb_ws[r], lc = lce_ws[r];
        svb += lv; sce += lc;
        sloss += (tt == 0) ? lc : (lv + LAMCE * lc);
        sce0  += (tt == 0) ? lc : 0.0f;
    }
    __shared__ float rv[256], rc[256], rl[256], r0[256];
    rv[tid] = svb; rc[tid] = sce; rl[tid] = sloss; r0[tid] = sce0;
    __syncthreads();
    if (tid < 32) {
        float a = 0.0f, b = 0.0f, c = 0.0f, d = 0.0f;
        #pragma unroll
        for (int w = 0; w < 8; ++w) {
            int k = tid + (w << 5);
            a += rv[k]; b += rc[k]; c += rl[k]; d += r0[k];
        }
        a = wave_sum32(a); b = wave_sum32(b); c = wave_sum32(c); d = wave_sum32(d);
        if (tid == 0) {
            int cnt0 = 0;
            for (int bb = 0; bb < 8; ++bb) cnt0 += (tarr[bb] == 0) ? 1 : 0;
            const float inv = 1.0f / (float)NROWS;
            float loss_mean = c * inv;
            float vb_mean   = a * inv;
            float ce_mean   = b * inv;
            float n0 = (float)cnt0 * 1024.0f;
            float l0_mean = (n0 > 0.0f) ? (d / fmaxf(n0, 1.0f)) : 0.0f;
            float l_T = vb_mean * (float)TSTEPS + l0_mean;
            float bpt = l_T / 0.69314718055994530942f;
            out5[0] = loss_mean; out5[1] = vb_mean; out5[2] = ce_mean;
            out5[3] = l_T;       out5[4] = bpt;
        }
    }
}

extern "C" void kernel_launch(void* const* d_in, const int* in_sizes, int n_in,
                              void* d_out, int out_size, void* d_ws, size_t ws_size,
                              hipStream_t stream) {
    const float* logits = (const float*)d_in[0];
    const int*   data   = (const int*)d_in[1];
    const int*   tarr   = (const int*)d_in[2];
    const int*   xtp1   = (const int*)d_in[3];
    float* out = (float*)d_out;
    float* lvb = (float*)d_ws;           // 8192 floats
    float* lce = lvb + NROWS;            // 8192 floats

    d3pm_row_kernel<<<NROWS, 256, 0, stream>>>(logits, data, tarr, xtp1, out, lvb, lce);
    d3pm_final_kernel<<<1, 256, 0, stream>>>(lvb, lce, tarr, out + (size_t)NROWS * KDIM);
}